// ClipSeemFusion_49598282334691
// MI455X (gfx1250) — compile-verified
//
#include <hip/hip_runtime.h>

#define NVX 64
#define NVOXELS (NVX*NVX*NVX)   // 262144
#define IMW 320
#define IMH 240
#define NFEAT 128
#define NCLS 143
#define TRUNC_INV (1.0f/0.12f)
#define WSREC_BYTES ((size_t)NVOXELS*32)

typedef float v2f __attribute__((ext_vector_type(2)));
typedef float v4f __attribute__((ext_vector_type(4)));
typedef float v8f __attribute__((ext_vector_type(8)));

__device__ __forceinline__ float unnormW(float g){ return ((g+1.0f)*(float)IMW - 1.0f)*0.5f; }
__device__ __forceinline__ float unnormH(float g){ return ((g+1.0f)*(float)IMH - 1.0f)*0.5f; }

// uvz = M*xyz - q  with M = K*R^T, q = M*t  (matches einsum('bji,bnj->bin'))
__device__ __forceinline__ void build_proj(const float* __restrict__ poses,
                                           const float* __restrict__ Km,
                                           int b, float M[3][3], float q[3]) {
  const float* P = poses + b*16;
  float t0 = P[3], t1 = P[7], t2 = P[11];
  #pragma unroll
  for (int i = 0; i < 3; ++i) {
    #pragma unroll
    for (int j = 0; j < 3; ++j)
      M[i][j] = Km[i*3+0]*P[j*4+0] + Km[i*3+1]*P[j*4+1] + Km[i*3+2]*P[j*4+2];
    q[i] = M[i][0]*t0 + M[i][1]*t1 + M[i][2]*t2;
  }
}

__device__ __forceinline__ void project(const float M[3][3], const float q[3],
                                        float wx, float wy, float wz,
                                        float& gx, float& gy, float& z) {
  float u = M[0][0]*wx + M[0][1]*wy + M[0][2]*wz - q[0];
  float v = M[1][0]*wx + M[1][1]*wy + M[1][2]*wz - q[1];
  z       = M[2][0]*wx + M[2][1]*wy + M[2][2]*wz - q[2];
  float zs = (z > 1e-6f) ? z : 1.0f;
  gx = (u/zs + 0.5f)*(2.0f/(float)IMW) - 1.0f;
  gy = (v/zs + 0.5f)*(2.0f/(float)IMH) - 1.0f;
}

__device__ __forceinline__ bool nearest_idx(float gx, float gy, int& off) {
  float x = unnormW(gx), y = unnormH(gy);
  float ix = floorf(x + 0.5f), iy = floorf(y + 0.5f);
  bool inb = (ix >= 0.f) && (ix <= (float)(IMW-1)) && (iy >= 0.f) && (iy <= (float)(IMH-1));
  int xc = (int)fminf(fmaxf(ix, 0.f), (float)(IMW-1));
  int yc = (int)fminf(fmaxf(iy, 0.f), (float)(IMH-1));
  off = yc*IMW + xc;
  return inb;
}

__device__ __forceinline__ void view_valid(const float* __restrict__ depth, int b,
                                           float gx, float gy, float z,
                                           bool& vm, bool& tv, float& tsdf) {
  int off; bool inb = nearest_idx(gx, gy, off);
  float dep = inb ? depth[b*IMH*IMW + off] : 0.0f;
  float sdf = (dep - z)*TRUNC_INV;
  tsdf = fminf(fmaxf(sdf, -1.0f), 1.0f);
  bool v_ = (fabsf(gx) <= 1.0f) && (fabsf(gy) <= 1.0f) && (z > 0.0f);
  vm = v_ && (fabsf(sdf) <= 1.0f);
  tv = v_ && (sdf > -1.0f);
}

// ---------------------------------------------------------------------------
// K1: tsdf / weights / rgb via V_WMMA_F32_16X16X4_F32 projection (one wave =
// 16 voxels; A = stacked per-view P matrices duplicated in rows 0-5 / 8-13,
// B = voxel homogeneous coords; lanes<16 handle view0, lanes>=16 view1,
// combined with shfl_xor(.,16)).  Optionally emits a 32B/voxel sampling
// record into d_ws so K2/K3 never redo projection/validity math:
//   f32 wx1_0, wy1_0, wx1_1, wy1_1
//   i32 xy0 (x0|y0<<16), xy1, labs (lab0|vm0<<15|lab1<<16|vm1<<31), nv
// ---------------------------------------------------------------------------
__global__ void __launch_bounds__(256) k_scalar_fuse(
    const float* __restrict__ depth, const float* __restrict__ rgbim,
    const float* __restrict__ poses, const float* __restrict__ Km,
    const int* __restrict__ pano,
    const float* __restrict__ tsdf0, const float* __restrict__ rgb0,
    const int* __restrict__ weight0, const int* __restrict__ tw0,
    float* __restrict__ fused, int* __restrict__ wout, int* __restrict__ twout,
    float* __restrict__ wsrec)
{
  int tid   = blockIdx.x*256 + threadIdx.x;
  int lane  = tid & 31;
  int col   = lane & 15;          // D-matrix column / voxel-in-wave
  int vb    = lane >> 4;          // view this lane processes
  int v     = (tid >> 5)*16 + col;

  float M[2][3][3], q[2][3];
  build_proj(poses, Km, 0, M[0], q[0]);
  build_proj(poses, Km, 1, M[1], q[1]);

  float wx = (float)(v >> 12)      * 0.04f - 1.28f;
  float wy = (float)((v >> 6)&63)  * 0.04f - 1.28f;
  float wz = (float)(v & 63)       * 0.04f + 0.2f;

  // A row supplied by this lane: row = lane&15 (rows 8..13 duplicate 0..5)
  int rr = col & 7;
  float a0=0.f, a1=0.f, a2=0.f, a3=0.f;
  #pragma unroll
  for (int rowc = 0; rowc < 6; ++rowc) {
    int vw = rowc/3, ri = rowc%3;
    bool sel = (rr == rowc);
    a0 = sel ? M[vw][ri][0] : a0;
    a1 = sel ? M[vw][ri][1] : a1;
    a2 = sel ? M[vw][ri][2] : a2;
    a3 = sel ? -q[vw][ri]   : a3;
  }
  v2f Amat, Bmat;
  Amat.x = vb ? a2 : a0;
  Amat.y = vb ? a3 : a1;
  Bmat.x = vb ? wz : wx;
  Bmat.y = vb ? 1.0f : wy;
  v8f Cz = {0.f,0.f,0.f,0.f,0.f,0.f,0.f,0.f};
  v8f D = __builtin_amdgcn_wmma_f32_16x16x4_f32(false, Amat, false, Bmat,
                                                (short)0, Cz, false, false);

  float u  = vb ? D[3] : D[0];
  float vv = vb ? D[4] : D[1];
  float z  = vb ? D[5] : D[2];
  float zs = (z > 1e-6f) ? z : 1.0f;
  float gx = (u/zs + 0.5f)*(2.0f/(float)IMW) - 1.0f;
  float gy = (vv/zs + 0.5f)*(2.0f/(float)IMH) - 1.0f;

  // nearest-sample: depth -> validity/tsdf, pano -> label
  int noff; bool ninb = nearest_idx(gx, gy, noff);
  float dep = ninb ? depth[vb*IMH*IMW + noff] : 0.0f;
  float sdf = (dep - z)*TRUNC_INV;
  float tsdfv = fminf(fmaxf(sdf, -1.0f), 1.0f);
  bool v_ = (fabsf(gx) <= 1.0f) && (fabsf(gy) <= 1.0f) && (z > 0.0f);
  bool vm = v_ && (fabsf(sdf) <= 1.0f);
  bool tv = v_ && (sdf > -1.0f);
  int lab = ninb ? pano[vb*IMH*IMW + noff] : 0;

  // bilinear setup (shared by rgb here and by K2 via the ws record)
  float x = unnormW(gx), y = unnormH(gy);
  float x0 = floorf(x), y0 = floorf(y);
  float wx1 = x - x0, wy1 = y - y0;

  // rgb bilinear, layout (B,H,W,3)
  float rs0 = 0.f, rs1 = 0.f, rs2 = 0.f;
  #pragma unroll
  for (int t = 0; t < 4; ++t) {
    int dx = t >> 1, dy = t & 1;
    float xi = x0 + (float)dx, yi = y0 + (float)dy;
    bool inb = (xi >= 0.f) && (xi <= (float)(IMW-1)) && (yi >= 0.f) && (yi <= (float)(IMH-1));
    float w = (dx ? wx1 : 1.0f - wx1)*(dy ? wy1 : 1.0f - wy1);
    float wm = inb ? w : 0.0f;
    int xc = (int)fminf(fmaxf(xi, 0.f), (float)(IMW-1));
    int yc = (int)fminf(fmaxf(yi, 0.f), (float)(IMH-1));
    const float* p = rgbim + ((size_t)vb*IMH*IMW + (size_t)yc*IMW + xc)*3;
    rs0 += p[0]*wm; rs1 += p[1]*wm; rs2 += p[2]*wm;
  }

  // pack this lane's sampling state for the ws record
  int x0i = (int)fminf(fmaxf(x0, -32768.f), 32767.f);
  int y0i = (int)fminf(fmaxf(y0, -32768.f), 32767.f);
  int xyp = (x0i & 0xffff) | (y0i << 16);
  int h   = (lab & 0xff) | ((int)vm << 15);

  // combine the two views across lane pairs (l <-> l+16); all shfls convergent
  float tc  = tv ? tsdfv : 0.0f;
  float tcs = tc + __shfl_xor(tc, 16);
  int   btw = (int)tv + __shfl_xor((int)tv, 16);
  int   nv  = (int)vm + __shfl_xor((int)vm, 16);
  float r0 = vm ? rs0 : 0.0f, r1 = vm ? rs1 : 0.0f, r2 = vm ? rs2 : 0.0f;
  r0 += __shfl_xor(r0, 16); r1 += __shfl_xor(r1, 16); r2 += __shfl_xor(r2, 16);
  float wx1_o = __shfl_xor(wx1, 16);
  float wy1_o = __shfl_xor(wy1, 16);
  int   xyp_o = __shfl_xor(xyp, 16);
  int   h_o   = __shfl_xor(h, 16);

  if (vb == 0) {
    int   tw0v = tw0[v];
    int   w0v  = weight0[v];
    float t0v  = tsdf0[v];
    int   ntw  = tw0v + btw;
    float out_t;
    if (btw > 0) { float dn = (float)ntw; out_t = tcs/dn + t0v*((float)tw0v/dn); }
    else           out_t = t0v;
    float* fp = fused + (size_t)v*132;
    fp[0] = out_t;
    float g0 = rgb0[v*3+0], g1 = rgb0[v*3+1], g2 = rgb0[v*3+2];
    if (nv > 0) {
      float inv = 1.0f/(float)(w0v + nv);   // running mean closed form
      fp[1] = (g0*(float)w0v + r0)*inv;
      fp[2] = (g1*(float)w0v + r1)*inv;
      fp[3] = (g2*(float)w0v + r2)*inv;
    } else { fp[1] = g0; fp[2] = g1; fp[3] = g2; }
    wout[v]  = w0v + nv;
    twout[v] = ntw;

    if (wsrec) {
      float* r = wsrec + (size_t)v*8;
      r[0] = wx1;  r[1] = wy1;   // view0
      r[2] = wx1_o; r[3] = wy1_o; // view1
      int* ri = (int*)r;
      ri[4] = xyp;
      ri[5] = xyp_o;
      ri[6] = (h & 0xffff) | (h_o << 16);
      ri[7] = nv;
    }
  }
}

// ---------------------------------------------------------------------------
// K2 (fast): CLIP fusion driven entirely by the ws record — no projection,
// no divides, no depth loads.  One wave per voxel, lane owns 4 channels.
// ---------------------------------------------------------------------------
__global__ void __launch_bounds__(256) k_clip_fast(
    const float* __restrict__ clipim, const float* __restrict__ clip0,
    const int* __restrict__ weight0, const float* __restrict__ wsrec,
    float* __restrict__ fused)
{
  int gtid = blockIdx.x*256 + threadIdx.x;
  int n    = gtid >> 5;
  int lane = gtid & 31;
  if (n >= NVOXELS) return;

  const float* rec = wsrec + (size_t)n*8;
  v4f wv = *(const v4f*)rec;              // wx1_0, wy1_0, wx1_1, wy1_1
  const int* ri = (const int*)(rec + 4);
  int xyp[2] = { ri[0], ri[1] };
  int labs = ri[2];
  int nv   = ri[3];

  int w0v = weight0[n];
  const v4f* c0p = (const v4f*)(clip0 + (size_t)n*NFEAT) + lane;
  v4f c0 = __builtin_nontemporal_load(c0p);
  __builtin_prefetch(clip0 + (size_t)(n + 8)*NFEAT + lane*4, 0, 1);

  v4f acc = c0 * (float)w0v;
  #pragma unroll
  for (int b = 0; b < 2; ++b) {
    bool vmb = (labs >> (15 + b*16)) & 1;
    if (vmb) {
      int x0 = (int)(short)(xyp[b] & 0xffff);
      int y0 = xyp[b] >> 16;
      float wx1 = b ? wv.z : wv.x;
      float wy1 = b ? wv.w : wv.y;
      #pragma unroll
      for (int t = 0; t < 4; ++t) {
        int dx = t >> 1, dy = t & 1;
        int xi = x0 + dx, yi = y0 + dy;
        bool inb = ((unsigned)xi < IMW) && ((unsigned)yi < IMH);
        float w = (dx ? wx1 : 1.0f - wx1)*(dy ? wy1 : 1.0f - wy1);
        float wm = inb ? w : 0.0f;
        int xc = min(max(xi, 0), IMW-1);
        int yc = min(max(yi, 0), IMH-1);
        const float* base = clipim + ((size_t)b*NFEAT + (size_t)lane*4)*(IMH*IMW)
                                   + (size_t)yc*IMW + xc;
        acc.x += base[0*IMH*IMW]*wm;
        acc.y += base[1*IMH*IMW]*wm;
        acc.z += base[2*IMH*IMW]*wm;
        acc.w += base[3*IMH*IMW]*wm;
      }
    }
  }
  v4f outv;
  if (nv > 0) { float inv = 1.0f/(float)(w0v + nv); outv = acc*inv; }
  else          outv = c0;
  v4f* op = (v4f*)(fused + (size_t)n*132 + 4) + lane;
  __builtin_nontemporal_store(outv, op);
}

// K2 (fallback): recomputes projection/validity when d_ws is too small.
__global__ void __launch_bounds__(256) k_clip_slow(
    const float* __restrict__ depth, const float* __restrict__ poses,
    const float* __restrict__ Km, const float* __restrict__ clipim,
    const float* __restrict__ clip0, const int* __restrict__ weight0,
    float* __restrict__ fused)
{
  int gtid = blockIdx.x*256 + threadIdx.x;
  int n    = gtid >> 5;
  int lane = gtid & 31;
  if (n >= NVOXELS) return;

  float M[2][3][3], q[2][3];
  build_proj(poses, Km, 0, M[0], q[0]);
  build_proj(poses, Km, 1, M[1], q[1]);

  float wxw = (float)(n >> 12)      * 0.04f - 1.28f;
  float wyw = (float)((n >> 6)&63)  * 0.04f - 1.28f;
  float wzw = (float)(n & 63)       * 0.04f + 0.2f;

  int w0v = weight0[n];
  const v4f* c0p = (const v4f*)(clip0 + (size_t)n*NFEAT) + lane;
  v4f c0 = __builtin_nontemporal_load(c0p);

  v4f acc = c0 * (float)w0v;
  int nv = 0;
  #pragma unroll
  for (int b = 0; b < 2; ++b) {
    float gx, gy, z;
    project(M[b], q[b], wxw, wyw, wzw, gx, gy, z);
    bool vm, tv; float ts;
    view_valid(depth, b, gx, gy, z, vm, tv, ts);
    if (vm) {
      ++nv;
      float x = unnormW(gx), y = unnormH(gy);
      float x0 = floorf(x), y0 = floorf(y);
      float wx1 = x - x0, wy1 = y - y0;
      #pragma unroll
      for (int t = 0; t < 4; ++t) {
        int dx = t >> 1, dy = t & 1;
        float xi = x0 + (float)dx, yi = y0 + (float)dy;
        bool inb = (xi >= 0.f) && (xi <= (float)(IMW-1)) && (yi >= 0.f) && (yi <= (float)(IMH-1));
        float w = (dx ? wx1 : 1.0f - wx1)*(dy ? wy1 : 1.0f - wy1);
        float wm = inb ? w : 0.0f;
        int xc = (int)fminf(fmaxf(xi, 0.f), (float)(IMW-1));
        int yc = (int)fminf(fmaxf(yi, 0.f), (float)(IMH-1));
        const float* base = clipim + ((size_t)b*NFEAT + (size_t)lane*4)*(IMH*IMW)
                                   + (size_t)yc*IMW + xc;
        acc.x += base[0*IMH*IMW]*wm;
        acc.y += base[1*IMH*IMW]*wm;
        acc.z += base[2*IMH*IMW]*wm;
        acc.w += base[3*IMH*IMW]*wm;
      }
    }
  }
  v4f outv;
  if (nv > 0) { float inv = 1.0f/(float)(w0v + nv); outv = acc*inv; }
  else          outv = c0;
  v4f* op = (v4f*)(fused + (size_t)n*132 + 4) + lane;
  __builtin_nontemporal_store(outv, op);
}

// ---------------------------------------------------------------------------
// K3 (fast): labels from the packed ws record; one broadcast int per voxel,
// perfectly coalesced 150MB write, essentially zero arithmetic.
// ---------------------------------------------------------------------------
__global__ void __launch_bounds__(256) k_labels_fast(
    const int* __restrict__ labels0, const float* __restrict__ wsrec,
    int* __restrict__ labout)
{
  int g = blockIdx.x*256 + threadIdx.x;
  if (g >= NVOXELS*NCLS) return;
  int n = g / NCLS;
  int c = g - n*NCLS;
  int labs = ((const int*)wsrec)[(size_t)n*8 + 6];
  int lab0 = labs & 0xff,         vm0 = (labs >> 15) & 1;
  int lab1 = (labs >> 16) & 0xff, vm1 = (labs >> 31) & 1;
  labout[g] = labels0[g] + ((vm0 && lab0 == c) ? 1 : 0)
                         + ((vm1 && lab1 == c) ? 1 : 0);
}

// K3 (fallback): full recompute when d_ws is too small.
__global__ void __launch_bounds__(256) k_labels_slow(
    const float* __restrict__ depth, const float* __restrict__ poses,
    const float* __restrict__ Km, const int* __restrict__ pano,
    const int* __restrict__ labels0, int* __restrict__ labout)
{
  int g = blockIdx.x*256 + threadIdx.x;
  if (g >= NVOXELS*NCLS) return;
  int n = g / NCLS;
  int c = g - n*NCLS;

  float M[2][3][3], q[2][3];
  build_proj(poses, Km, 0, M[0], q[0]);
  build_proj(poses, Km, 1, M[1], q[1]);

  float wxw = (float)(n >> 12)      * 0.04f - 1.28f;
  float wyw = (float)((n >> 6)&63)  * 0.04f - 1.28f;
  float wzw = (float)(n & 63)       * 0.04f + 0.2f;

  int val = labels0[g];
  #pragma unroll
  for (int b = 0; b < 2; ++b) {
    float gx, gy, z;
    project(M[b], q[b], wxw, wyw, wzw, gx, gy, z);
    bool vm, tv; float ts;
    view_valid(depth, b, gx, gy, z, vm, tv, ts);
    int off; bool inb = nearest_idx(gx, gy, off);
    int lab = inb ? pano[b*IMH*IMW + off] : 0;
    val += (vm && (lab == c)) ? 1 : 0;
  }
  labout[g] = val;
}

extern "C" void kernel_launch(void* const* d_in, const int* in_sizes, int n_in,
                              void* d_out, int out_size, void* d_ws, size_t ws_size,
                              hipStream_t stream) {
  (void)in_sizes; (void)n_in; (void)out_size;
  const float* depth   = (const float*)d_in[0];
  const float* rgbim   = (const float*)d_in[1];
  const float* poses   = (const float*)d_in[2];
  const float* Km      = (const float*)d_in[3];
  const float* clipim  = (const float*)d_in[4];
  const int*   pano    = (const int*)d_in[5];
  const float* tsdf0   = (const float*)d_in[6];
  const float* rgb0    = (const float*)d_in[7];
  const float* clip0   = (const float*)d_in[8];
  const int*   weight0 = (const int*)d_in[9];
  const int*   tw0     = (const int*)d_in[10];
  const int*   labels0 = (const int*)d_in[11];

  float* fused = (float*)d_out;
  int*   wout  = (int*)(fused + (size_t)NVOXELS*132);
  int*   twout = wout + NVOXELS;
  int*   labout= twout + NVOXELS;

  // Deterministic host-side choice: ws_size is fixed for the life of the run.
  bool ws_ok = (d_ws != nullptr) && (ws_size >= WSREC_BYTES);
  float* wsrec = ws_ok ? (float*)d_ws : nullptr;

  // K1: 16 voxels/wave (WMMA) -> 128 voxels per 256-thread block
  k_scalar_fuse<<<NVOXELS/128, 256, 0, stream>>>(depth, rgbim, poses, Km, pano,
                                                 tsdf0, rgb0, weight0, tw0,
                                                 fused, wout, twout, wsrec);
  if (ws_ok) {
    k_clip_fast<<<NVOXELS/8, 256, 0, stream>>>(clipim, clip0, weight0, wsrec, fused);
    int totalL = NVOXELS*NCLS;
    k_labels_fast<<<(totalL + 255)/256, 256, 0, stream>>>(labels0, wsrec, labout);
  } else {
    k_clip_slow<<<NVOXELS/8, 256, 0, stream>>>(depth, poses, Km, clipim, clip0,
                                               weight0, fused);
    int totalL = NVOXELS*NCLS;
    k_labels_slow<<<(totalL + 255)/256, 256, 0, stream>>>(depth, poses, Km, pano,
                                                          labels0, labout);
  }
}